// SeesawLoss_48524540510729
// MI455X (gfx1250) — compile-verified
//
#include <hip/hip_runtime.h>
#include <hip/hip_bf16.h>

// ---------------------------------------------------------------------------
// Seesaw loss, MI455X (gfx1250).
// Memory-bound (128MB single read ~5.5us @ 23.3TB/s) + transcendental-heavy.
// No GEMM structure -> no WMMA; the CDNA5 win is the async global->LDS DMA
// path (ASYNCcnt) double-buffering rows, and wave32 shuffle reductions.
// ---------------------------------------------------------------------------

namespace {

constexpr int   kB      = 8192;
constexpr int   kC      = 4096;
constexpr float kP      = 0.8f;
constexpr float kQ      = 2.0f;
constexpr float kLogEps = -4.6051702f;   // log(0.01)

constexpr int TPB = 256;                 // 8 waves (wave32)
constexpr int VPT = kC / TPB;            // 16 floats per thread
constexpr int CH  = VPT / 4;             // 4 float4 chunks per thread
constexpr int NWG = 1024;
constexpr int RPW = kB / NWG;            // 8 rows per workgroup

// CDNA5 async global->LDS copy (VGLOBAL, tracked with ASYNCcnt).
// %0 = VGPR holding wave-relative LDS byte address, %1 = 64-bit global addr.
#define ASYNC_LOAD_B128(ldsoff, gptr)                                        \
  asm volatile("global_load_async_to_lds_b128 %0, %1, off"                   \
               :: "v"(ldsoff), "v"(gptr) : "memory")

#define S_WAIT_ASYNC(n) asm volatile("s_wait_asynccnt " #n ::: "memory")

// Issue 4 x b128 async loads covering this thread's 16 columns of one row.
// Chunk k covers cols [k*1024 + 4*tid, +4): per wave-instruction the 32 lanes
// touch a contiguous 512B span (perfectly coalesced DMA requests).
__device__ __forceinline__ void issue_row_copy(const float* __restrict__ g,
                                               const float* lds_base, int tid) {
#pragma unroll
  for (int k = 0; k < CH; ++k) {
    const int c = k * (TPB * 4) + tid * 4;
    const unsigned loff = (unsigned)(size_t)(const void*)(lds_base + c);
    ASYNC_LOAD_B128(loff, g + c);
  }
}

// Deterministic block-wide reduction (wave32 shuffles + 8-slot LDS scratch).
template <bool IS_MAX>
__device__ __forceinline__ float block_reduce(float v, float* scratch) {
#pragma unroll
  for (int off = 16; off > 0; off >>= 1) {
    const float o = __shfl_xor(v, off, 32);
    v = IS_MAX ? fmaxf(v, o) : (v + o);
  }
  const int lane = threadIdx.x & 31;
  const int wave = threadIdx.x >> 5;
  if (lane == 0) scratch[wave] = v;
  __syncthreads();
  if (wave == 0) {
    float x = (lane < TPB / 32) ? scratch[lane]
                                : (IS_MAX ? -__builtin_inff() : 0.0f);
#pragma unroll
    for (int off = 4; off > 0; off >>= 1) {
      const float o = __shfl_xor(x, off, 32);
      x = IS_MAX ? fmaxf(x, o) : (x + o);
    }
    if (lane == 0) scratch[0] = x;
  }
  __syncthreads();
  const float r = scratch[0];
  __syncthreads();  // scratch reusable afterwards
  return r;
}

// --------------------------- helper kernels --------------------------------

__global__ void zero_acc_kernel(unsigned* acc) {
  const int i = blockIdx.x * blockDim.x + threadIdx.x;
  if (i < kC) acc[i] = 0u;
}

__global__ void hist_kernel(const int* __restrict__ tgt, unsigned* acc) {
  const int i = blockIdx.x * blockDim.x + threadIdx.x;
  if (i < kB) atomicAdd(&acc[tgt[i]], 1u);
}

__global__ void log_acc_kernel(const unsigned* __restrict__ acc,
                               float* __restrict__ la) {
  const int i = blockIdx.x * blockDim.x + threadIdx.x;
  if (i < kC) la[i] = __logf((float)(acc[i] < 1u ? 1u : acc[i]));
}

// ----------------------------- main kernel ---------------------------------

__global__ void __launch_bounds__(TPB)
seesaw_main_kernel(const float* __restrict__ outs, const int* __restrict__ tgt,
                   const float* __restrict__ la_g, float* __restrict__ nll) {
  __shared__ float s_buf0[kC];       // 16KB row staging (ping)
  __shared__ float s_buf1[kC];       // 16KB row staging (pong)
  __shared__ float s_la[kC];         // 16KB log-count table
  __shared__ float s_red[TPB / 32];

  const int tid  = threadIdx.x;
  const int row0 = blockIdx.x * RPW;

  // Stage log-count table + first two rows via the async DMA path.
  issue_row_copy(la_g, s_la, tid);                                  // cnt +4
  issue_row_copy(outs + (size_t)row0 * kC, s_buf0, tid);            // cnt +4
  issue_row_copy(outs + (size_t)(row0 + 1) * kC, s_buf1, tid);      // cnt +4

  for (int i = 0; i < RPW; ++i) {
    float* cur = (i & 1) ? s_buf1 : s_buf0;

    // In-order async completion: retiring down to 4 outstanding guarantees
    // everything except the in-flight prefetch (next row) has landed.
    if (i == RPW - 1) { S_WAIT_ASYNC(0); } else { S_WAIT_ASYNC(4); }
    __syncthreads();  // whole buffer (all waves' DMA) now valid

    const int   row = row0 + i;
    const int   t   = tgt[row];
    const float xt  = cur[t];
    const float lat = s_la[t];

    float4 x[CH];
    float4 lav[CH];
#pragma unroll
    for (int k = 0; k < CH; ++k) {
      x[k]   = *(const float4*)(cur  + k * (TPB * 4) + tid * 4);
      lav[k] = *(const float4*)(s_la + k * (TPB * 4) + tid * 4);
    }

    // ---- pass 1: logsumexp of raw row (register-resident) ----
    float m = -__builtin_inff();
#pragma unroll
    for (int k = 0; k < CH; ++k) {
      m = fmaxf(m, fmaxf(fmaxf(x[k].x, x[k].y), fmaxf(x[k].z, x[k].w)));
    }
    m = block_reduce<true>(m, s_red);
    float s = 0.0f;
#pragma unroll
    for (int k = 0; k < CH; ++k) {
      s += __expf(x[k].x - m) + __expf(x[k].y - m) +
           __expf(x[k].z - m) + __expf(x[k].w - m);
    }
    s = block_reduce<false>(s, s_red);
    const float lse0  = m + __logf(s);
    const float lclip = fmaxf(xt - lse0, kLogEps);  // consumes xt before reissue

    // All reads of `cur` are behind the reduction barriers -> safe to start
    // streaming row i+2 into this buffer while we do pass 2.
    if (i + 2 < RPW)
      issue_row_copy(outs + (size_t)(row + 2) * kC, cur, tid);

    // ---- pass 2: seesaw-adjusted logits; term vanishes at j==t by algebra ----
    float lg[VPT];
    float m1 = -__builtin_inff();
#pragma unroll
    for (int k = 0; k < CH; ++k) {
      const float* xe = (const float*)&x[k];
      const float* le = (const float*)&lav[k];
#pragma unroll
      for (int e = 0; e < 4; ++e) {
        const float mit  = kP * fminf(0.0f, le[e] - lat);
        const float comp = kQ * fmaxf(0.0f, (xe[e] - lse0) - lclip);
        const float v    = xe[e] + mit + comp;
        lg[k * 4 + e] = v;
        m1 = fmaxf(m1, v);
      }
    }
    m1 = block_reduce<true>(m1, s_red);
    float s1 = 0.0f;
#pragma unroll
    for (int j = 0; j < VPT; ++j) s1 += __expf(lg[j] - m1);
    s1 = block_reduce<false>(s1, s_red);

    if (tid == 0) nll[row] = (m1 + __logf(s1)) - xt;
  }
}

__global__ void mean_kernel(const float* __restrict__ nll, float* __restrict__ out) {
  __shared__ float s_red[TPB / 32];
  float s = 0.0f;
  for (int i = threadIdx.x; i < kB; i += TPB) s += nll[i];
  s = block_reduce<false>(s, s_red);
  if (threadIdx.x == 0) out[0] = s * (1.0f / (float)kB);
}

}  // namespace

// ---------------------------------------------------------------------------

extern "C" void kernel_launch(void* const* d_in, const int* in_sizes, int n_in,
                              void* d_out, int out_size, void* d_ws, size_t ws_size,
                              hipStream_t stream) {
  const float* outs = (const float*)d_in[0];   // [8192, 4096] f32
  const int*   tgt  = (const int*)d_in[1];     // [8192] i32

  // workspace layout: acc[C] u32 | la[C] f32 | nll[B] f32   (64 KB total)
  unsigned* acc = (unsigned*)d_ws;
  float*    la  = (float*)((char*)d_ws + (size_t)kC * sizeof(unsigned));
  float*    nll = (float*)((char*)d_ws + (size_t)2 * kC * sizeof(unsigned));

  zero_acc_kernel<<<kC / TPB, TPB, 0, stream>>>(acc);
  hist_kernel<<<kB / TPB, TPB, 0, stream>>>(tgt, acc);
  log_acc_kernel<<<kC / TPB, TPB, 0, stream>>>(acc, la);
  seesaw_main_kernel<<<NWG, TPB, 0, stream>>>(outs, tgt, la, nll);
  mean_kernel<<<1, TPB, 0, stream>>>(nll, (float*)d_out);
}